// MultiTurnCRF_69982197121446
// MI455X (gfx1250) — compile-verified
//
#include <hip/hip_runtime.h>
#include <math.h>

// ---------------------------------------------------------------------------
// MultiTurnCRF forward on MI455X (gfx1250, wave32, WMMA).
//
// The 131071-step log-semiring scan is lifted into ordinary f32 matrix
// products with per-strip rescaling: 512 parallel chunks x 256 sequential
// 128x128x128 matmuls on V_WMMA_F32_16X16X4_F32, then a small sequential
// combine kernel folds alpha through the 512 chunk products in log space.
//
// This revision packs both LDS operand matrices in a "K-packed per half-wave"
// layout so each ds_load_b128 delivers a lane's A/B operands for TWO K-slices,
// cutting DS instruction issue per step from ~352 to ~208 (DS issue is the
// limiter, not LDS bytes), while keeping the double-buffered software
// pipeline that already overlaps loads with WMMAs.
// ---------------------------------------------------------------------------

typedef float v2f __attribute__((ext_vector_type(2)));
typedef float v4f __attribute__((ext_vector_type(4)));
typedef float v8f __attribute__((ext_vector_type(8)));

#define SEQ_LEN 131072
#define NTAGS   128
#define NSTEPS  (SEQ_LEN - 1)   // transitions t = 1 .. NSTEPS
#define NCHUNK  512
#define CHUNK   256             // steps per chunk (NCHUNK*CHUNK == NSTEPS+1)

// K-packed layout: element (row/col, k) lives in region hi=(k>>1)&1 at packed
// index kp = 2*(k>>2) + (k&1). A b128 at kp=4*m yields K-slices 2m and 2m+1
// for that lane. Pitch 68 (mod 64 == 4) spreads the 16 active rows across
// distinct LDS bank groups; all b128 addresses stay 16-byte aligned.
#define KP 68
#define ETP_SEL    (2 * 128 * KP)      // floats per transition variant
#define ETP_FLOATS (3 * ETP_SEL)
#define PP_FLOATS  (2 * 128 * KP)
#define LDS_FLOATS (ETP_FLOATS + PP_FLOATS + CHUNK)

#define GOLD_BLOCKS 256

// ---------------------------------------------------------------------------
// Kernel 1: per-chunk product P_c = prod_{t in chunk} exp(trans_t)·diag(exp(em_t))
// One workgroup (8 waves) per chunk. Wave w owns row strip [16w, 16w+16).
// ---------------------------------------------------------------------------
__global__ __launch_bounds__(256)
void crf_chunk_kernel(const float* __restrict__ em,
                      const int*   __restrict__ who2who,
                      const float* __restrict__ trans_g,
                      const float* __restrict__ trans_u,
                      const float* __restrict__ trans_s,
                      float* __restrict__ Pout,   // [NCHUNK][128][128]
                      float* __restrict__ LSout)  // [NCHUNK][8] per-strip log scale
{
    extern __shared__ float smem[];
    float* ETP = smem;                      // exp(trans)^T, K-packed
    float* Pp  = smem + ETP_FLOATS;         // running product, K-packed
    int*   SEL = (int*)(smem + ETP_FLOATS + PP_FLOATS);

    const int tid   = threadIdx.x;
    const int lane  = tid & 31;
    const int wv    = tid >> 5;       // wave index 0..7 = row strip
    const int c     = blockIdx.x;
    const int lan15 = lane & 15;
    const int hi    = lane >> 4;      // 0 for lanes 0-15, 1 for lanes 16-31

    const int t0 = c * CHUNK + 1;
    const int t1 = ((c + 1) * CHUNK < NSTEPS) ? (c + 1) * CHUNK : NSTEPS; // inclusive

    // ---- build exp(trans)^T in K-packed layout ----
    for (int idx = tid; idx < 3 * 128 * 128; idx += 256) {
        const int s   = idx >> 14;
        const int k   = (idx >> 7) & 127;   // prev-tag (K index of the matmul)
        const int col = idx & 127;          // cur-tag  (N index)
        const float* T = (s == 0) ? trans_g : (s == 1) ? trans_u : trans_s;
        const int khi = (k >> 1) & 1;
        const int kp  = 2 * (k >> 2) + (k & 1);
        ETP[s * ETP_SEL + khi * (128 * KP) + col * KP + kp] = __expf(T[k * 128 + col]);
    }
    // ---- P = identity (K-packed; here "k" is the column index) ----
    for (int idx = tid; idx < 128 * 128; idx += 256) {
        const int r = idx >> 7, col = idx & 127;
        const int chi = (col >> 1) & 1;
        const int kp  = 2 * (col >> 2) + (col & 1);
        Pp[chi * (128 * KP) + r * KP + kp] = (r == col) ? 1.0f : 0.0f;
    }
    // ---- per-step transition selector table (kills per-step L2 latency) ----
    {
        const int t = t0 + tid;           // CHUNK == blockDim.x
        if (tid < CHUNK) {
            int sel = 0;
            if (t <= NSTEPS) {
                const int w2 = who2who[t];
                sel = (w2 == 0) ? 1 : ((w2 == 1) ? 2 : 0);
            }
            SEL[tid] = sel;
        }
    }
    __syncthreads();

    float ls = 0.0f;                  // per-strip log scale
    const int arow = 16 * wv + lan15; // row this lane supplies for the A operand

    // A operand base: this lane's row in its hi region of packed P.
    const float* abase = Pp + hi * (128 * KP) + arow * KP;
    // B operand bases per output tile (sel offset added per step).
    int colbase[8];
    #pragma unroll
    for (int j = 0; j < 8; ++j)
        colbase[j] = hi * (128 * KP) + (16 * j + lan15) * KP;
    // Store mapping: element (row, col) -> packed region by col.
    float* pstore = Pp + ((lan15 >> 1) & 1) * (128 * KP)
                       + (2 * (lan15 >> 2) + (lan15 & 1));

    // Prefetch first step's emission row.
    float eraw[8];
    {
        const float* emt = em + (size_t)t0 * NTAGS;
        #pragma unroll
        for (int j = 0; j < 8; ++j) eraw[j] = emt[16 * j + lan15];
    }

    for (int t = t0; t <= t1; ++t) {
        // prefetch next step's emission row (overlaps with the matmul below)
        float enext[8];
        {
            const int tn = (t < t1) ? t + 1 : t1;
            const float* emt = em + (size_t)tn * NTAGS;
            #pragma unroll
            for (int j = 0; j < 8; ++j) enext[j] = emt[16 * j + lan15];
        }

        const float* Eb = ETP + SEL[t - t0] * ETP_SEL;

        v8f acc[8];
        #pragma unroll
        for (int j = 0; j < 8; ++j) {
            v8f z = {0.f, 0.f, 0.f, 0.f, 0.f, 0.f, 0.f, 0.f};
            acc[j] = z;
        }

        // ---- P_new strip = P_old strip (16x128) x E (128x128) ----
        // Each b128 fetch covers two K-slices; double-buffered so the next
        // slice-pair's 9 loads overlap the current pair's 16 WMMAs.
        v4f ab[2], bb[2][8];
        ab[0] = *(const v4f*)(abase);
        #pragma unroll
        for (int j = 0; j < 8; ++j)
            bb[0][j] = *(const v4f*)(Eb + colbase[j]);

        #pragma unroll 2
        for (int kp2 = 0; kp2 < 16; ++kp2) {
            const int cur = kp2 & 1, nxt = cur ^ 1;
            if (kp2 < 15) {
                const int off = 4 * (kp2 + 1);
                ab[nxt] = *(const v4f*)(abase + off);
                #pragma unroll
                for (int j = 0; j < 8; ++j)
                    bb[nxt][j] = *(const v4f*)(Eb + colbase[j] + off);
            }
            const v2f alo = __builtin_shufflevector(ab[cur], ab[cur], 0, 1);
            const v2f ahi = __builtin_shufflevector(ab[cur], ab[cur], 2, 3);
            #pragma unroll
            for (int j = 0; j < 8; ++j) {
                const v2f blo = __builtin_shufflevector(bb[cur][j], bb[cur][j], 0, 1);
                acc[j] = __builtin_amdgcn_wmma_f32_16x16x4_f32(
                    false, alo, false, blo, (short)0, acc[j], false, false);
            }
            #pragma unroll
            for (int j = 0; j < 8; ++j) {
                const v2f bhi = __builtin_shufflevector(bb[cur][j], bb[cur][j], 2, 3);
                acc[j] = __builtin_amdgcn_wmma_f32_16x16x4_f32(
                    false, ahi, false, bhi, (short)0, acc[j], false, false);
            }
        }

        // apply emission column scaling; find strip max for renormalization
        float mx = 0.0f;
        #pragma unroll
        for (int j = 0; j < 8; ++j) {
            const float e = __expf(eraw[j]);
            acc[j] = acc[j] * e;
            #pragma unroll
            for (int q = 0; q < 8; ++q) mx = fmaxf(mx, acc[j][q]);
        }
        #pragma unroll
        for (int off = 16; off > 0; off >>= 1)
            mx = fmaxf(mx, __shfl_xor(mx, off, 32));
        const float inv = 1.0f / mx;
        ls += __logf(mx);
        #pragma unroll
        for (int j = 0; j < 8; ++j) acc[j] = acc[j] * inv;

        // C/D layout -> K-packed LDS strip (only this wave touches these rows,
        // per-wave LDS ops are in order -> no barrier needed).
        #pragma unroll
        for (int j = 0; j < 8; ++j) {
            #pragma unroll
            for (int q = 0; q < 8; ++q) {
                const int row = 16 * wv + q + 8 * hi;
                pstore[row * KP + 8 * j] = acc[j][q];
            }
        }

        #pragma unroll
        for (int j = 0; j < 8; ++j) eraw[j] = enext[j];
    }

    // write per-strip log scale
    if (lane == 0) LSout[c * 8 + wv] = ls;

    // write chunk product (row-major) to global scratch
    __syncthreads();
    float* Pg = Pout + (size_t)c * 128 * 128;
    for (int idx = tid; idx < 128 * 128; idx += 256) {
        const int r = idx >> 7, col = idx & 127;
        const int chi = (col >> 1) & 1;
        const int kp  = 2 * (col >> 2) + (col & 1);
        Pg[idx] = Pp[chi * (128 * KP) + r * KP + kp];
    }
}

// ---------------------------------------------------------------------------
// Kernel 2: fold alpha0 = em[0] through the 512 chunk products in log space,
// then total = logsumexp(alpha). Single block, thread j owns alpha[j].
// ---------------------------------------------------------------------------
__global__ __launch_bounds__(128)
void crf_combine_kernel(const float* __restrict__ em,
                        const float* __restrict__ Pin,
                        const float* __restrict__ LSin,
                        float* __restrict__ out)
{
    __shared__ float sh_ea[128];
    __shared__ float sh_red[128];
    const int j = threadIdx.x;

    float alpha = em[j];  // alpha0 = emissions[0]

    for (int c = 0; c < NCHUNK; ++c) {
        // a_i = alpha[i] + logscale of row-strip(i) for this chunk
        const float a = alpha + LSin[c * 8 + (j >> 4)];
        sh_red[j] = a;
        __syncthreads();
        for (int off = 64; off > 0; off >>= 1) {
            if (j < off) sh_red[j] = fmaxf(sh_red[j], sh_red[j + off]);
            __syncthreads();
        }
        const float m = sh_red[0];
        __syncthreads();
        sh_ea[j] = __expf(a - m);
        __syncthreads();

        const float* Pc = Pin + (size_t)c * 128 * 128;
        // 4 partial sums break the FMA dependence chain so loads pipeline.
        float s0 = 0.f, s1 = 0.f, s2 = 0.f, s3 = 0.f;
        #pragma unroll 4
        for (int i = 0; i < 128; i += 4) {
            s0 = fmaf(sh_ea[i + 0], Pc[(i + 0) * 128 + j], s0);
            s1 = fmaf(sh_ea[i + 1], Pc[(i + 1) * 128 + j], s1);
            s2 = fmaf(sh_ea[i + 2], Pc[(i + 2) * 128 + j], s2);
            s3 = fmaf(sh_ea[i + 3], Pc[(i + 3) * 128 + j], s3);
        }
        alpha = m + __logf((s0 + s1) + (s2 + s3));
        __syncthreads();
    }

    // total = logsumexp(alpha)
    sh_red[j] = alpha;
    __syncthreads();
    for (int off = 64; off > 0; off >>= 1) {
        if (j < off) sh_red[j] = fmaxf(sh_red[j], sh_red[j + off]);
        __syncthreads();
    }
    const float m = sh_red[0];
    __syncthreads();
    sh_red[j] = __expf(alpha - m);
    __syncthreads();
    for (int off = 64; off > 0; off >>= 1) {
        if (j < off) sh_red[j] += sh_red[j + off];
        __syncthreads();
    }
    if (j == 0) out[1] = m + __logf(sh_red[0]);
}

// ---------------------------------------------------------------------------
// Gold score: two-stage deterministic reduction (no float atomics).
// ---------------------------------------------------------------------------
__global__ __launch_bounds__(256)
void gold_partial_kernel(const float* __restrict__ em,
                         const int*   __restrict__ tags,
                         const int*   __restrict__ who2who,
                         const float* __restrict__ trans_g,
                         const float* __restrict__ trans_u,
                         const float* __restrict__ trans_s,
                         float* __restrict__ partial)
{
    __shared__ float red[256];
    const int tid = threadIdx.x;
    float s = 0.0f;
    for (int t = blockIdx.x * 256 + tid; t < SEQ_LEN; t += 256 * GOLD_BLOCKS) {
        const int cur = tags[t];
        s += em[(size_t)t * NTAGS + cur];
        if (t > 0) {
            const int w2  = who2who[t];
            const int sel = (w2 == 0) ? 1 : ((w2 == 1) ? 2 : 0);
            const float* T = (sel == 0) ? trans_g : (sel == 1) ? trans_u : trans_s;
            s += T[tags[t - 1] * NTAGS + cur];
        }
    }
    red[tid] = s;
    __syncthreads();
    for (int off = 128; off > 0; off >>= 1) {
        if (tid < off) red[tid] += red[tid + off];
        __syncthreads();
    }
    if (tid == 0) partial[blockIdx.x] = red[0];
}

__global__ __launch_bounds__(32)
void gold_final_kernel(const float* __restrict__ partial, float* __restrict__ out)
{
    if (threadIdx.x == 0) {
        float s = 0.0f;
        for (int i = 0; i < GOLD_BLOCKS; ++i) s += partial[i];  // fixed order
        out[0] = s;
    }
}

// ---------------------------------------------------------------------------
extern "C" void kernel_launch(void* const* d_in, const int* in_sizes, int n_in,
                              void* d_out, int out_size, void* d_ws, size_t ws_size,
                              hipStream_t stream) {
    const float* em      = (const float*)d_in[0];
    const int*   tags    = (const int*)d_in[1];
    const int*   who2who = (const int*)d_in[2];
    const float* trans_g = (const float*)d_in[3];
    const float* trans_u = (const float*)d_in[4];
    const float* trans_s = (const float*)d_in[5];
    float* out = (float*)d_out;

    float* P       = (float*)d_ws;                       // NCHUNK*128*128 floats (32 MB)
    float* LS      = P + (size_t)NCHUNK * 128 * 128;     // NCHUNK*8 floats
    float* partial = LS + (size_t)NCHUNK * 8;            // GOLD_BLOCKS floats

    const size_t lds_bytes = (size_t)LDS_FLOATS * sizeof(float);

    crf_chunk_kernel<<<NCHUNK, 256, lds_bytes, stream>>>(
        em, who2who, trans_g, trans_u, trans_s, P, LS);
    crf_combine_kernel<<<1, 128, 0, stream>>>(em, P, LS, out);
    gold_partial_kernel<<<GOLD_BLOCKS, 256, 0, stream>>>(
        em, tags, who2who, trans_g, trans_u, trans_s, partial);
    gold_final_kernel<<<1, 32, 0, stream>>>(partial, out);
}